// RoPECrossAttention_11269994185101
// MI455X (gfx1250) — compile-verified
//
#include <hip/hip_runtime.h>
#include <hip/hip_bf16.h>
#include <math.h>

typedef __attribute__((ext_vector_type(2))) float v2f;
typedef __attribute__((ext_vector_type(4))) float v4f;
typedef __attribute__((ext_vector_type(8))) float v8f;
typedef int b128i __attribute__((vector_size(16)));   // matches builtin param type

// D = A(16x4) * B(4x16) + C, fp32 WMMA
#define WMMA_F32_4(a, b, c) \
  __builtin_amdgcn_wmma_f32_16x16x4_f32(false, (a), false, (b), (short)0, (c), false, false)

constexpr int BB = 4;
constexpr int LL = 2048;   // Lt == Ls
constexpr int DD = 1024;
constexpr int HH = 16;
constexpr int DH = 64;

// ---------------------------------------------------------------------------
// CDNA5 async global->LDS staging (ASYNCcnt-tracked); sync fallback if the
// builtins are not available in this toolchain.
// ---------------------------------------------------------------------------
#if defined(__gfx1250__) && __has_builtin(__builtin_amdgcn_global_load_async_to_lds_b128)
#define USE_ASYNC_LDS 1
#endif

__device__ __forceinline__ void g2l_b128(const float* __restrict__ g, float* l) {
#ifdef USE_ASYNC_LDS
  __builtin_amdgcn_global_load_async_to_lds_b128(
      (__attribute__((address_space(1))) b128i*)g,
      (__attribute__((address_space(3))) b128i*)l, 0, 0);
#else
  *(v4f*)l = *(const v4f*)g;
#endif
}

#if defined(USE_ASYNC_LDS) && __has_builtin(__builtin_amdgcn_s_wait_asynccnt)
#define WAIT_ASYNC(n) __builtin_amdgcn_s_wait_asynccnt(n)
#else
#define WAIT_ASYNC(n)
#endif

// ---------------------------------------------------------------------------
// C[M,N] = A[M,K] * W[N,K]^T
// Block: 128 threads (4 waves) computes a 64x64 tile; wave w owns rows
// [w*16, w*16+16) x all 64 cols.  K is consumed in 16-wide panels that are
// double-buffered in LDS via async global->LDS loads (coalesced b128),
// issued one stage ahead of the WMMA consumption.
// LDS rows padded to stride 20 floats -> conflict-free 16-lane b64 reads.
// ---------------------------------------------------------------------------
constexpr int KB = 16;              // k-panel depth
constexpr int LSTR = 20;            // padded LDS row stride (floats)
constexpr int ASYNC_PER_STAGE = 4;  // b128 async issues per thread per stage

__global__ void gemm_nt_wmma(const float* __restrict__ A, const float* __restrict__ W,
                             float* __restrict__ C, int M, int N, int K) {
  __shared__ float As[2][64][LSTR];
  __shared__ float Ws[2][64][LSTR];

  const int tid  = threadIdx.x;
  const int lane = tid & 31;
  const int wave = tid >> 5;
  const int lr   = lane & 15;
  const int kh   = (lane >> 4) * 2;
  const int m0b  = blockIdx.x * 64;
  const int n0b  = blockIdx.y * 64;

  v8f acc[4] = {};

  // stage(buf, k0): copy A[m0b..+64, k0..+16] and W[n0b..+64, k0..+16] to LDS.
  // 64 rows x 16 floats = 256 b128 slots per tile; 2 slots/thread/tile.
  auto stage = [&](int buf, int k0) {
#pragma unroll
    for (int rep = 0; rep < 2; ++rep) {
      int slot = tid + rep * 128;
      int row  = slot >> 2;
      int c4   = (slot & 3) * 4;
      g2l_b128(A + (size_t)(m0b + row) * K + k0 + c4, &As[buf][row][c4]);
      g2l_b128(W + (size_t)(n0b + row) * K + k0 + c4, &Ws[buf][row][c4]);
    }
  };

  stage(0, 0);
  int buf = 0;
  for (int k0 = 0; k0 < K; k0 += KB) {
    const bool more = (k0 + KB) < K;
    if (more) {
      stage(buf ^ 1, k0 + KB);
      WAIT_ASYNC(ASYNC_PER_STAGE);   // current panel done; next may be in flight
    } else {
      WAIT_ASYNC(0);
    }
    __syncthreads();

#pragma unroll
    for (int kk = 0; kk < KB; kk += 4) {
      v2f a = *(const v2f*)&As[buf][wave * 16 + lr][kk + kh];
#pragma unroll
      for (int t = 0; t < 4; ++t) {
        v2f b = *(const v2f*)&Ws[buf][t * 16 + lr][kk + kh];
        acc[t] = WMMA_F32_4(a, b, acc[t]);
      }
    }
    __syncthreads();   // panel consumed by all waves; safe to overwrite
    buf ^= 1;
  }

  const int mb = m0b + wave * 16 + ((lane >> 4) * 8);
#pragma unroll
  for (int t = 0; t < 4; ++t)
#pragma unroll
    for (int i = 0; i < 8; ++i)
      C[(size_t)(mb + i) * N + n0b + t * 16 + lr] = acc[t][i];
}

// ---------------------------------------------------------------------------
// In-place RoPE on [B, L, H, DH] fp32; one thread per (b,l,h,i<32) pair.
// ---------------------------------------------------------------------------
__global__ void rope_inplace(float* __restrict__ t, int total) {
  int idx = blockIdx.x * blockDim.x + threadIdx.x;
  if (idx >= total) return;
  const int half = DH / 2;                 // 32
  int i    = idx & (half - 1);
  int rest = idx >> 5;                     // (b*L + l)*H + h
  int h    = rest & (HH - 1);
  int bl   = rest >> 4;                    // b*L + l
  int l    = bl & (LL - 1);

  float inv = expf(-(float)i * (9.210340371976184f / 32.0f));  // 10000^(-i/32)
  float ang = (float)l * inv;
  float c = cosf(ang);
  float s = sinf(ang);

  size_t base = (size_t)bl * DD + (size_t)h * DH;
  float t1 = t[base + i];
  float t2 = t[base + half + i];
  t[base + i]        = t1 * c - t2 * s;
  t[base + half + i] = t1 * s + t2 * c;
}

// ---------------------------------------------------------------------------
// Attention for one (b, h, 16-row q block).  block = 128 (4 waves).
// Full 16 x 2048 score strip lives in LDS (320KB/WGP on CDNA5).
// Phases 1 & 3 double-buffer their WMMA fragments in registers so loads for
// the next tile are in flight while the current tile's WMMAs execute.
// ---------------------------------------------------------------------------
__global__ void attn_kernel(const float* __restrict__ q, const float* __restrict__ k,
                            const float* __restrict__ v,
                            const unsigned char* __restrict__ kpm,
                            float* __restrict__ out) {
  extern __shared__ float smem[];
  const int SROW = LL + 4;                 // stride 2052 -> conflict-free rows
  float* S      = smem;                    // [16][SROW]
  float* pmax   = S + 16 * SROW;           // [16][8]
  float* psum   = pmax + 128;              // [16][8]
  float* rowmax = psum + 128;              // [16]
  float* rowinv = rowmax + 16;             // [16]

  const int lane = threadIdx.x & 31;
  const int wave = threadIdx.x >> 5;
  const int lr   = lane & 15;
  const int kh   = (lane >> 4) * 2;
  const int q0   = blockIdx.x * 16;
  const int h    = blockIdx.y;
  const int b    = blockIdx.z;

  const size_t bhBase = (size_t)b * LL * DD + (size_t)h * DH;

  // ---- Phase 1: S = (Q K^T) * scale, masked, into LDS -------------------
  v2f aq[16];
  {
    const float* qrow = q + bhBase + (size_t)(q0 + lr) * DD + kh;
#pragma unroll
    for (int s = 0; s < 16; ++s) aq[s] = *(const v2f*)(qrow + s * 4);
  }
  const float scale = 0.125f;              // 1/sqrt(64)
  const int JT = LL / 16;                  // 128 j-tiles

  v2f bcur[16], bnxt[16];
  {
    const float* krow = k + bhBase + (size_t)(wave * 16 + lr) * DD + kh;
#pragma unroll
    for (int s = 0; s < 16; ++s) bcur[s] = *(const v2f*)(krow + s * 4);
  }
  for (int jt = wave; jt < JT; jt += 4) {
    const int jn = jt + 4;
    if (jn < JT) {
      const float* krow = k + bhBase + (size_t)(jn * 16 + lr) * DD + kh;
#pragma unroll
      for (int s = 0; s < 16; ++s) bnxt[s] = *(const v2f*)(krow + s * 4);
    }
    v8f c = {};
#pragma unroll
    for (int s = 0; s < 16; ++s) c = WMMA_F32_4(aq[s], bcur[s], c);

    const int j = jt * 16 + lr;
    const bool masked = kpm[(size_t)b * LL + j] != 0;
    const int mb = (lane >> 4) * 8;
#pragma unroll
    for (int i = 0; i < 8; ++i) {
      float val = masked ? -INFINITY : c[i] * scale;
      S[(mb + i) * SROW + j] = val;
    }
#pragma unroll
    for (int s = 0; s < 16; ++s) bcur[s] = bnxt[s];
  }
  __syncthreads();

  // ---- Phase 2: row softmax in LDS (row = tid&15, chunk of 256 = tid>>4) -
  {
    const int r  = threadIdx.x & 15;
    const int ch = threadIdx.x >> 4;
    float* srow = S + r * SROW + ch * 256;
    float m = -INFINITY;
    for (int j = 0; j < 256; ++j) m = fmaxf(m, srow[j]);
    pmax[r * 8 + ch] = m;
    __syncthreads();
    if (threadIdx.x < 16) {
      float mm = -INFINITY;
      for (int cc = 0; cc < 8; ++cc) mm = fmaxf(mm, pmax[threadIdx.x * 8 + cc]);
      rowmax[threadIdx.x] = mm;
    }
    __syncthreads();
    float rm  = rowmax[r];
    float sum = 0.0f;
    for (int j = 0; j < 256; ++j) {
      float e = expf(srow[j] - rm);
      srow[j] = e;
      sum += e;
    }
    psum[r * 8 + ch] = sum;
    __syncthreads();
    if (threadIdx.x < 16) {
      float ss = 0.0f;
      for (int cc = 0; cc < 8; ++cc) ss += psum[threadIdx.x * 8 + cc];
      rowinv[threadIdx.x] = 1.0f / ss;
    }
    __syncthreads();
  }

  // ---- Phase 3: O = P * V; each wave one 16-wide d tile ------------------
  {
    const int d0 = wave * 16;
    v8f c = {};
    const float* vcol = v + bhBase + d0 + lr;
    const float* sa = S + lr * SROW + kh;

    v2f acur[4], bcur2[4], anxt[4], bnxt2[4];
#pragma unroll
    for (int u = 0; u < 4; ++u) {
      acur[u] = *(const v2f*)(sa + u * 4);
      bcur2[u].x = vcol[(size_t)(u * 4 + kh) * DD];
      bcur2[u].y = vcol[(size_t)(u * 4 + kh + 1) * DD];
    }
    for (int j0 = 0; j0 < LL; j0 += 16) {
      const int jn = j0 + 16;
      if (jn < LL) {
#pragma unroll
        for (int u = 0; u < 4; ++u) {
          anxt[u] = *(const v2f*)(sa + jn + u * 4);
          bnxt2[u].x = vcol[(size_t)(jn + u * 4 + kh) * DD];
          bnxt2[u].y = vcol[(size_t)(jn + u * 4 + kh + 1) * DD];
        }
      }
#pragma unroll
      for (int u = 0; u < 4; ++u) c = WMMA_F32_4(acur[u], bcur2[u], c);
#pragma unroll
      for (int u = 0; u < 4; ++u) { acur[u] = anxt[u]; bcur2[u] = bnxt2[u]; }
    }

    const int mb = (lane >> 4) * 8;
#pragma unroll
    for (int i = 0; i < 8; ++i) {
      float inv = rowinv[mb + i];
      out[bhBase + (size_t)(q0 + mb + i) * DD + d0 + lr] = c[i] * inv;
    }
  }
}

// ---------------------------------------------------------------------------
extern "C" void kernel_launch(void* const* d_in, const int* in_sizes, int n_in,
                              void* d_out, int out_size, void* d_ws, size_t ws_size,
                              hipStream_t stream) {
  (void)in_sizes; (void)n_in; (void)out_size; (void)ws_size;

  const float* x   = (const float*)d_in[0];
  const float* ctx = (const float*)d_in[1];
  const unsigned char* kpm = (const unsigned char*)d_in[2];
  const float* Wq  = (const float*)d_in[3];
  const float* Wk  = (const float*)d_in[4];
  const float* Wv  = (const float*)d_in[5];
  const float* Wo  = (const float*)d_in[6];
  float* outp = (float*)d_out;

  const size_t tensElems = (size_t)BB * LL * DD;   // 8.4M floats each
  float* qb = (float*)d_ws;
  float* kb = qb + tensElems;
  float* vb = kb + tensElems;
  float* ao = vb + tensElems;

  const int M = BB * LL;                           // 8192
  dim3 ggrid(M / 64, DD / 64);                     // (128, 16)

  gemm_nt_wmma<<<ggrid, 128, 0, stream>>>(x,   Wq, qb, M, DD, DD);
  gemm_nt_wmma<<<ggrid, 128, 0, stream>>>(ctx, Wk, kb, M, DD, DD);
  gemm_nt_wmma<<<ggrid, 128, 0, stream>>>(ctx, Wv, vb, M, DD, DD);

  int npairs = BB * LL * HH * (DH / 2);            // 4,194,304
  rope_inplace<<<npairs / 256, 256, 0, stream>>>(qb, npairs);
  rope_inplace<<<npairs / 256, 256, 0, stream>>>(kb, npairs);

  size_t shmem = (size_t)(16 * (LL + 4) + 128 + 128 + 16 + 16) * sizeof(float);
  (void)hipFuncSetAttribute((const void*)attn_kernel,
                            hipFuncAttributeMaxDynamicSharedMemorySize, (int)shmem);
  attn_kernel<<<dim3(LL / 16, HH, BB), 128, shmem, stream>>>(qb, kb, vb, kpm, ao);

  gemm_nt_wmma<<<ggrid, 128, 0, stream>>>(ao, Wo, outp, M, DD, DD);
}